// MutualInformationLoss_24464133718342
// MI455X (gfx1250) — compile-verified
//
#include <hip/hip_runtime.h>
#include <hip/hip_bf16.h>

// Mutual-information loss for MI455X (gfx1250).
// Exponent tiles computed with V_WMMA_F32_16X16X4_F32 via the Gram trick,
// pre-scaled into the log2 domain so WMMA output feeds v_exp_f32 directly:
//   log2-exponent_a(m,n) = u'_m u'_n + q'_m + q'_n,
//   u' = (u/sigma)*sqrt(log2e), q' = -(u/sigma)^2/2*log2e.
// Joint kernel value = exp2(ea)*exp2(eb)  (fma instead of a third exp).
// Per 16x16 tile per wave: 2 loads, 2 wmma, 16 v_exp_f32, ~14 VALU ops.
// Column loop split into 8 chunks -> 10016 independent waves; partial row
// sums land in private workspace slices (deterministic, no float atomics).
// Wave coordinates go through readfirstlane so loop control is pure SALU
// and EXEC is statically all-ones across every WMMA.

#define MI_SIGMA    0.4f
#define MI_INVSIG   (1.0f / MI_SIGMA)
#define NSAMP       10000
#define NTILE       626             // 10000/16 + 1 (one pad tile)
#define NPAD        (NTILE * 16)    // 10016
#define NCHUNK      8
#define CHUNK_TILES 79              // ceil(626/8)
#define EPSV        1e-10f
#define PADU        1.0e4f          // pad: exp2 of ~-7e7 underflows to 0.0f
#define LOG2E_F     1.44269504088896340736f
#define SQRT_L2E_F  1.20112240872420969798f  // sqrt(log2(e))

typedef float v2f __attribute__((ext_vector_type(2)));
typedef float v8f __attribute__((ext_vector_type(8)));

// ---------------------------------------------------------------- prep ----
// prep layout (floats), per set s in {0,1}, base = s*4*NPAD:
//   [0*NPAD) u' = (f[idx]/sigma)*sqrt(log2e)    [1*NPAD) v' likewise
//   [2*NPAD) qa' = -(f[idx]/sigma)^2/2*log2e    [3*NPAD) qb' likewise
__global__ void __launch_bounds__(256)
mi_prep_kernel(const float* __restrict__ f,
               const float* __restrict__ s1,
               const float* __restrict__ s2,
               const int* __restrict__ idx1,
               const int* __restrict__ idx2,
               float* __restrict__ prep) {
  int i = blockIdx.x * blockDim.x + threadIdx.x;
  if (i >= 2 * NPAD) return;
  const int set = i / NPAD;
  const int k   = i - set * NPAD;
  const int*   idx = set ? idx2 : idx1;
  const float* src = set ? s2 : s1;
  float* base = prep + set * 4 * NPAD;
  float u, v;
  if (k < NSAMP) {
    const int j = idx[k];
    u = f[j]   * MI_INVSIG;
    v = src[j] * MI_INVSIG;
  } else {
    u = PADU; v = PADU;   // padded samples contribute exp2(-huge) == 0
  }
  base[0 * NPAD + k] = u * SQRT_L2E_F;
  base[1 * NPAD + k] = v * SQRT_L2E_F;
  base[2 * NPAD + k] = -0.5f * u * u * LOG2E_F;
  base[3 * NPAD + k] = -0.5f * v * v * LOG2E_F;
}

// -------------------------------------------------------------- rowsum ----
// Wave = (set, rowTile t, column chunk c). Partial sums land in
//   P + ((set*NCHUNK + c)*3)*NPAD : Sa[NPAD], Sb[NPAD], Sab[NPAD].
__global__ void __launch_bounds__(256)
mi_rowsum_kernel(const float* __restrict__ prep, float* __restrict__ P) {
  const int lane = threadIdx.x & 31;
  // Wave coordinates: force into SGPRs so loop bounds/control are scalar.
  const int gwave = __builtin_amdgcn_readfirstlane(
      blockIdx.x * (blockDim.x >> 5) + (threadIdx.x >> 5));
  if (gwave >= 2 * NTILE * NCHUNK) return;   // whole waves exit: EXEC all-1s
  const int set   = gwave / (NTILE * NCHUNK);
  const int rem   = gwave - set * (NTILE * NCHUNK);
  const int t     = rem / NCHUNK;
  const int chunk = rem - t * NCHUNK;

  const float* u  = prep + set * 4 * NPAD;
  const float* v  = u + 1 * NPAD;
  const float* qa = u + 2 * NPAD;
  const float* qb = u + 3 * NPAD;

  const int  lane15 = lane & 15;
  const bool hi     = (lane >= 16);          // A frag: lanes>=16 hold K=2,3
  const int  mrow   = t * 16 + lane15;

  // A fragments: row m = [u'_m, qa'_m, 1, 0]  (marginal-a), analog for b.
  v2f Aa, Ab;
  if (!hi) { Aa[0] = u[mrow]; Aa[1] = qa[mrow]; Ab[0] = v[mrow]; Ab[1] = qb[mrow]; }
  else     { Aa[0] = 1.0f;    Aa[1] = 0.0f;     Ab[0] = 1.0f;    Ab[1] = 0.0f;    }

  const int c0 = chunk * CHUNK_TILES;                               // SGPR
  const int c1 = (c0 + CHUNK_TILES < NTILE) ? (c0 + CHUNK_TILES) : NTILE;

  // Per-lane column pointers, advanced by one tile (16 floats) per step.
  // B fragments: col n = [u'_n, 1, qa'_n, 0] -> lanes<16 read u'/v',
  // lanes>=16 read qa'/qb'; second component is a lane constant.
  const float* pBa = (hi ? qa : u) + (c0 * 16 + lane15);
  const float* pBb = (hi ? qb : v) + (c0 * 16 + lane15);
  const float  Bc1 = hi ? 0.0f : 1.0f;

  v8f accA = {}, accB = {}, accAB = {};

  for (int ct = c0; ct < c1; ++ct) {
    v2f Ba, Bb;
    Ba[0] = *pBa; Ba[1] = Bc1;
    Bb[0] = *pBb; Bb[1] = Bc1;
    pBa += 16; pBb += 16;

    v8f zero = {};
    v8f ea = __builtin_amdgcn_wmma_f32_16x16x4_f32(
        false, Aa, false, Ba, (short)0, zero, false, false);
    v8f eb = __builtin_amdgcn_wmma_f32_16x16x4_f32(
        false, Ab, false, Bb, (short)0, zero, false, false);

#pragma unroll
    for (int r = 0; r < 8; ++r) {
      const float va = __builtin_amdgcn_exp2f(ea[r]);   // exp2: log2 domain
      const float vb = __builtin_amdgcn_exp2f(eb[r]);
      accA[r]  += va;
      accB[r]  += vb;
      accAB[r]  = __builtin_fmaf(va, vb, accAB[r]);     // exp(ea+eb)=va*vb
    }
  }

  float* Pa  = P + ((set * NCHUNK + chunk) * 3) * NPAD;
  float* Pb  = Pa + 1 * NPAD;
  float* Pab = Pa + 2 * NPAD;

  // C/D layout: VGPR r, lanes 0-15 -> M=r ; lanes 16-31 -> M=r+8.
  // XOR-reduce within each 16-lane half (masks <= 8 never cross halves).
#pragma unroll
  for (int r = 0; r < 8; ++r) {
    float sa = accA[r], sb = accB[r], sab = accAB[r];
#pragma unroll
    for (int m = 1; m <= 8; m <<= 1) {
      sa  += __shfl_xor(sa,  m, 32);
      sb  += __shfl_xor(sb,  m, 32);
      sab += __shfl_xor(sab, m, 32);
    }
    if ((lane & 15) == 0) {
      const int row = t * 16 + r + (hi ? 8 : 0);
      Pa[row] = sa; Pb[row] = sb; Pab[row] = sab;
    }
  }
}

// ------------------------------------------------------------ finalize ----
__device__ __forceinline__ float mi_block_reduce(float x, float* red) {
  const int tid = threadIdx.x;
  red[tid] = x;
  __syncthreads();
  for (int s = 128; s > 0; s >>= 1) {
    if (tid < s) red[tid] += red[tid + s];
    __syncthreads();
  }
  const float r = red[0];
  __syncthreads();
  return r;
}

__global__ void __launch_bounds__(256)
mi_finalize_kernel(const float* __restrict__ P, float* __restrict__ out) {
  __shared__ float red[256];
  const int tid = threadIdx.x;
  const float invN = 1.0f / (float)NSAMP;
  float mi_total = 0.0f;

  for (int set = 0; set < 2; ++set) {
    const float* base = P + (set * NCHUNK) * 3 * NPAD;

    float ta = 0.0f, tb = 0.0f, tab = 0.0f;
    for (int i = tid; i < NSAMP; i += 256) {
      float sa = 0.0f, sb = 0.0f, sab = 0.0f;
#pragma unroll
      for (int c = 0; c < NCHUNK; ++c) {
        const float* Pc = base + c * 3 * NPAD;
        sa  += Pc[0 * NPAD + i];
        sb  += Pc[1 * NPAD + i];
        sab += Pc[2 * NPAD + i];
      }
      ta += sa; tb += sb; tab += sab;
    }
    const float TA  = mi_block_reduce(ta,  red);
    const float TB  = mi_block_reduce(tb,  red);
    const float TAB = mi_block_reduce(tab, red);

    // reference: p = (S/N) / (sum(S/N) + eps)
    const float da  = TA  * invN + EPSV;
    const float db  = TB  * invN + EPSV;
    const float dab = TAB * invN + EPSV;

    float mi = 0.0f;
    for (int i = tid; i < NSAMP; i += 256) {
      float sa = 0.0f, sb = 0.0f, sab = 0.0f;
#pragma unroll
      for (int c = 0; c < NCHUNK; ++c) {
        const float* Pc = base + c * 3 * NPAD;
        sa  += Pc[0 * NPAD + i];
        sb  += Pc[1 * NPAD + i];
        sab += Pc[2 * NPAD + i];
      }
      const float pa  = (sa  * invN) / da;
      const float pb  = (sb  * invN) / db;
      const float pab = (sab * invN) / dab;
      mi += pab * __logf((pab + EPSV) / (pa * pb + EPSV));
    }
    mi_total += mi_block_reduce(mi, red);
  }

  if (tid == 0) out[0] = -mi_total;
}

// -------------------------------------------------------------- launch ----
extern "C" void kernel_launch(void* const* d_in, const int* in_sizes, int n_in,
                              void* d_out, int out_size, void* d_ws, size_t ws_size,
                              hipStream_t stream) {
  const float* f    = (const float*)d_in[0];
  const float* s1   = (const float*)d_in[1];
  const float* s2   = (const float*)d_in[2];
  const int*   idx1 = (const int*)d_in[3];
  const int*   idx2 = (const int*)d_in[4];

  float* prep = (float*)d_ws;             // 2 * 4 * NPAD floats  (~320 KB)
  float* P    = prep + 2 * 4 * NPAD;      // 2 * NCHUNK * 3 * NPAD (~1.9 MB)

  const int prepElems = 2 * NPAD;
  mi_prep_kernel<<<(prepElems + 255) / 256, 256, 0, stream>>>(
      f, s1, s2, idx1, idx2, prep);

  const int totalWaves = 2 * NTILE * NCHUNK;   // 10016 waves
  mi_rowsum_kernel<<<(totalWaves + 7) / 8, 256, 0, stream>>>(prep, P);

  mi_finalize_kernel<<<1, 256, 0, stream>>>(P, (float*)d_out);
}